// ZCANormSVDPIv3_37151467110668
// MI455X (gfx1250) — compile-verified
//
#include <hip/hip_runtime.h>

typedef __attribute__((ext_vector_type(2))) float v2f;
typedef __attribute__((ext_vector_type(8))) float v8f;

#define BATCH 64
#define C_DIM 64
#define M_DIM 8192
#define EPSV  1e-5f
#define KCH   256
#define NT    4      // n-tiles (64 cols each) per block in k_zgemm

__device__ __forceinline__ v8f wmma_f32_k4(v2f a, v2f b, v8f c) {
  // D = A(16x4,f32) * B(4x16,f32) + C(16x16,f32)
  return __builtin_amdgcn_wmma_f32_16x16x4_f32(false, a, false, b, (short)0, c,
                                               false, false);
}

// Async global->LDS copy, 16 bytes per lane. ASYNCcnt-tracked (gfx1250).
// VDST = per-lane LDS byte address (low 32 bits of a generic LDS pointer are
// the wave-relative LDS offset), VADDR = per-lane 64-bit global address.
__device__ __forceinline__ void async_copy_b128(uint32_t lds_off,
                                                const void* gaddr) {
  asm volatile("global_load_async_to_lds_b128 %0, %1, off"
               :: "v"(lds_off), "v"(gaddr)
               : "memory");
}

template <int N>
__device__ __forceinline__ void wait_asynccnt() {
  asm volatile("s_wait_asynccnt %0" :: "i"(N) : "memory");
}

// ---------------------------------------------------------------------------
// Kernel 1: per-batch Gram via WMMA with double-buffered async LDS fills,
//           row sums, shrinkage -> g_cov, g_mu
// ---------------------------------------------------------------------------
__global__ __launch_bounds__(256)
void k_cov(const float* __restrict__ x, float* __restrict__ g_cov,
           float* __restrict__ g_mu) {
  __shared__ float xs[2][C_DIM * KCH];   // 2 x 64 KB ping-pong
  __shared__ float rowsum[C_DIM];
  __shared__ float redbuf[2];            // trace, second_trace
  __shared__ float sc[2];                // rho, fhat
  const int b = blockIdx.x;
  const int t = threadIdx.x;
  const int lane = t & 31;
  const int w = t >> 5;
  const size_t xbase = (size_t)b * C_DIM * M_DIM;

  if (t < C_DIM) rowsum[t] = 0.0f;
  if (t == 0) { redbuf[0] = 0.0f; redbuf[1] = 0.0f; }

  // issue async fill of one 64xKCH chunk into buffer `buf` (16 B128/wave)
  auto issue = [&](int ch, int buf) {
#pragma unroll
    for (int i = 0; i < 16; ++i) {
      int f = i * 256 + t;               // float4 index within chunk
      int row = f >> 6;
      int c4 = f & 63;
      async_copy_b128((uint32_t)(uintptr_t)&xs[buf][row * KCH + c4 * 4],
                      x + xbase + (size_t)row * M_DIM + (size_t)ch * KCH +
                          c4 * 4);
    }
  };
  issue(0, 0);

  float rs[16];
#pragma unroll
  for (int i = 0; i < 16; ++i) rs[i] = 0.0f;

  // two 16x16 output tiles per wave: (ti, tj0) and (ti, tj1)
  const int ti = w >> 1;
  const int tj0 = (w * 2) & 3;
  const int tj1 = (w * 2 + 1) & 3;
  const int mrow = lane & 15;
  const int kb = (lane >> 4) * 2;
  v8f acc0 = {}; v8f acc1 = {};

  const int NCH = M_DIM / KCH;
  for (int ch = 0; ch < NCH; ++ch) {
    const int cur = ch & 1;
    if (ch + 1 < NCH) {
      issue(ch + 1, cur ^ 1);            // overlap next fill with this compute
      wait_asynccnt<16>();               // in-order: current chunk landed
    } else {
      wait_asynccnt<0>();
    }
    __syncthreads();
    const float* X = xs[cur];
    // fold row sums from LDS (overlaps with WMMA issue)
#pragma unroll
    for (int i = 0; i < 16; ++i) {
      int f = i * 256 + t;
      int row = f >> 6;
      int c4 = f & 63;
      const float4 v = *(const float4*)&X[row * KCH + c4 * 4];
      rs[i] += v.x + v.y + v.z + v.w;
    }
    // Gram accumulation, K in steps of 4 (A operand shared by both tiles)
    for (int k = 0; k < KCH; k += 4) {
      v2f a, b0, b1;
      a.x  = X[(ti  * 16 + mrow) * KCH + k + kb];
      a.y  = X[(ti  * 16 + mrow) * KCH + k + kb + 1];
      b0.x = X[(tj0 * 16 + mrow) * KCH + k + kb];
      b0.y = X[(tj0 * 16 + mrow) * KCH + k + kb + 1];
      b1.x = X[(tj1 * 16 + mrow) * KCH + k + kb];
      b1.y = X[(tj1 * 16 + mrow) * KCH + k + kb + 1];
      acc0 = wmma_f32_k4(a, b0, acc0);
      acc1 = wmma_f32_k4(a, b1, acc1);
    }
    __syncthreads();                     // release buffer `cur` for refill
  }

  // stage Gram tiles into LDS (reuse xs[0] as 64x64), C/D layout:
  // VGPR r -> row r (lanes 0-15) / row r+8 (lanes 16-31)
  float* gram = xs[0];
  {
    const int mo = (lane >> 4) * 8;
#pragma unroll
    for (int r = 0; r < 8; ++r) {
      gram[(ti * 16 + r + mo) * 64 + tj0 * 16 + mrow] = acc0[r];
      gram[(ti * 16 + r + mo) * 64 + tj1 * 16 + mrow] = acc1[r];
    }
  }
#pragma unroll
  for (int i = 0; i < 16; ++i) atomicAdd(&rowsum[i * 4 + (t >> 6)], rs[i]);
  __syncthreads();

  // cov = gram/M - mu mu^T ; accumulate trace & Frobenius statistics
  const float invM = 1.0f / (float)M_DIM;
  float ltr = 0.0f, lst = 0.0f;
#pragma unroll
  for (int i = 0; i < 16; ++i) {
    int e = i * 256 + t;
    int r = e >> 6, cc = e & 63;
    float cov = gram[e] * invM - (rowsum[r] * invM) * (rowsum[cc] * invM);
    gram[e] = cov;
    lst += cov * cov;
    if (r == cc) ltr += cov;
  }
  atomicAdd(&redbuf[0], ltr);
  atomicAdd(&redbuf[1], lst);
  if (t < C_DIM) g_mu[b * C_DIM + t] = rowsum[t] * invM;
  __syncthreads();
  if (t == 0) {
    float tr = redbuf[0], st = redbuf[1];
    float m = (float)M_DIM, c = (float)C_DIM;
    float num = (m - 2.0f) / m * st + tr * tr;
    float den = (m + 2.0f) * (st - tr * tr / c);
    sc[0] = fminf(num / den, 1.0f);      // rho
    sc[1] = tr / c;                      // fhat
  }
  __syncthreads();
  const float rho = sc[0], fhat = sc[1];
#pragma unroll
  for (int i = 0; i < 16; ++i) {
    int e = i * 256 + t;
    int r = e >> 6, cc = e & 63;
    float diag = (r == cc) ? 1.0f : 0.0f;
    g_cov[(size_t)b * 4096 + e] =
        rho * fhat * diag + (1.0f - rho) * gram[e] + EPSV * diag;
  }
}

// ---------------------------------------------------------------------------
// Kernel 2: per-batch cyclic Jacobi eigensolver (SPD 64x64) -> g_U, g_L desc
// ---------------------------------------------------------------------------
__global__ __launch_bounds__(64)
void k_eig(const float* __restrict__ g_cov, float* __restrict__ g_U,
           float* __restrict__ g_L) {
  __shared__ float A_[64 * 64], V_[64 * 64];
  __shared__ float cs[2];
  const int b = blockIdx.x, t = threadIdx.x;
  for (int i = 0; i < 64; ++i) {
    A_[i * 64 + t] = g_cov[(size_t)b * 4096 + i * 64 + t];
    V_[i * 64 + t] = (i == t) ? 1.0f : 0.0f;
  }
  __syncthreads();
  for (int sweep = 0; sweep < 8; ++sweep) {
    for (int p = 0; p < 63; ++p) {
      for (int q = p + 1; q < 64; ++q) {
        if (t == 0) {
          float apq = A_[p * 64 + q];
          float c = 1.0f, s = 0.0f;
          if (fabsf(apq) > 1e-12f) {
            float theta = (A_[q * 64 + q] - A_[p * 64 + p]) / (2.0f * apq);
            float sg = theta >= 0.0f ? 1.0f : -1.0f;
            float tt = sg / (fabsf(theta) + sqrtf(theta * theta + 1.0f));
            c = 1.0f / sqrtf(tt * tt + 1.0f);
            s = tt * c;
          }
          cs[0] = c; cs[1] = s;
        }
        __syncthreads();
        const float c = cs[0], s = cs[1];
        // column rotation of A and V
        float aip = A_[t * 64 + p], aiq = A_[t * 64 + q];
        A_[t * 64 + p] = c * aip - s * aiq;
        A_[t * 64 + q] = s * aip + c * aiq;
        float vip = V_[t * 64 + p], viq = V_[t * 64 + q];
        V_[t * 64 + p] = c * vip - s * viq;
        V_[t * 64 + q] = s * vip + c * viq;
        __syncthreads();
        // row rotation of A
        float api = A_[p * 64 + t], aqi = A_[q * 64 + t];
        A_[p * 64 + t] = c * api - s * aqi;
        A_[q * 64 + t] = s * api + c * aqi;
        __syncthreads();
      }
    }
  }
  // sort descending via rank, scatter eigenpairs
  float li = A_[t * 64 + t];
  int rank = 0;
  for (int j = 0; j < 64; ++j) {
    float lj = A_[j * 64 + j];
    rank += (lj > li) || (lj == li && j < t);
  }
  g_L[b * 64 + rank] = li;
  for (int r = 0; r < 64; ++r)
    g_U[(size_t)b * 4096 + r * 64 + rank] = V_[r * 64 + t];
}

// ---------------------------------------------------------------------------
// Kernel 3: cross-batch counter reduction
// ---------------------------------------------------------------------------
__global__ __launch_bounds__(64)
void k_counter(const float* __restrict__ g_L, int* __restrict__ g_counter) {
  __shared__ int conds[64];
  const int j = threadIdx.x;
  int cond = 0;
  for (int b = 0; b < 64; ++b) {
    float total = 0.0f, prefix = 0.0f;
    for (int k = 0; k < 64; ++k) {
      float v = g_L[b * 64 + k];
      total += v;
      if (k <= j) prefix += v;
    }
    float lj = g_L[b * 64 + j];
    if (prefix / total >= 1.0f - EPSV || lj <= EPSV) cond = 1;
  }
  conds[j] = cond;
  __syncthreads();
  if (j == 0) {
    int cnt = 64;
    for (int k = 0; k < 64; ++k)
      if (conds[k]) { cnt = k; break; }
    *g_counter = cnt;
  }
}

// ---------------------------------------------------------------------------
// Kernel 4: per-batch deflation scan -> S_hat, and t = S_hat @ mu
// ---------------------------------------------------------------------------
__global__ __launch_bounds__(64)
void k_deflate(const float* __restrict__ g_cov, const float* __restrict__ g_U,
               const float* __restrict__ g_L, const int* __restrict__ g_counter,
               const float* __restrict__ g_mu, float* __restrict__ g_S,
               float* __restrict__ g_t) {
  __shared__ float Cc_[64 * 64], S_[64 * 64];
  __shared__ float v_[64], w_[64], prodv[64], prodw[64];
  __shared__ float invl;
  __shared__ int okf;
  const int b = blockIdx.x, t = threadIdx.x;
  const int counter = *g_counter;
  for (int i = 0; i < 64; ++i) {
    Cc_[i * 64 + t] = g_cov[(size_t)b * 4096 + i * 64 + t];
    S_[i * 64 + t] = 0.0f;
  }
  __syncthreads();
  int active = 1;
  for (int j = 0; j < 64; ++j) {
    v_[t] = g_U[(size_t)b * 4096 + t * 64 + j];
    __syncthreads();
    float acc = 0.0f;
    for (int k = 0; k < 64; ++k) acc += Cc_[t * 64 + k] * v_[k];
    w_[t] = acc;                      // w = cov_c @ v
    prodv[t] = v_[t] * v_[t];
    prodw[t] = v_[t] * acc;
    __syncthreads();
    if (t == 0) {
      float vv = 0.0f, vw = 0.0f;
      for (int k = 0; k < 64; ++k) { vv += prodv[k]; vw += prodw[k]; }
      float lam = vw / vv;
      float Lj = g_L[b * 64 + j];
      int ok = active && (j < counter) && !(lam < 0.0f) &&
               !(fabsf(lam - Lj) / Lj > 0.1f);
      okf = ok;
      invl = ok ? rsqrtf(lam) : 0.0f;
    }
    __syncthreads();
    const int ok = okf;
    if (ok) {
      const float inv = invl;
      const float vt = v_[t], wt = w_[t];
      for (int k = 0; k < 64; ++k) {
        S_[t * 64 + k] += inv * vt * v_[k];
        Cc_[t * 64 + k] -= wt * v_[k];   // cov_c -= (cov_c v) v^T
      }
    }
    active = ok;
    __syncthreads();
  }
  float accm = 0.0f;
  for (int k = 0; k < 64; ++k) {
    g_S[(size_t)b * 4096 + t * 64 + k] = S_[t * 64 + k];
    accm += S_[t * 64 + k] * g_mu[b * 64 + k];
  }
  g_t[b * 64 + t] = accm;              // (S_hat @ mu)[t]
}

// ---------------------------------------------------------------------------
// Kernel 5: Z = S_hat @ x - (S_hat @ mu) broadcast, WMMA + async x-tiles
// ---------------------------------------------------------------------------
__global__ __launch_bounds__(256)
void k_zgemm(const float* __restrict__ x, const float* __restrict__ g_S,
             const float* __restrict__ g_t, float* __restrict__ out) {
  __shared__ float Sl[64 * 64];        // S_hat, row-major [m][k], resident
  __shared__ float Xl[2][64 * 64];     // x tiles [k][n], ping-pong
  __shared__ float tl[64];             // S_hat @ mu, resident
  const int b = blockIdx.y;
  const int nbase = blockIdx.x * (64 * NT);
  const int t = threadIdx.x;
  const int lane = t & 31;
  const int w = t >> 5;
  const size_t xbase = (size_t)b * C_DIM * M_DIM;

  auto issueX = [&](int tile, int buf) {
    const int n0 = nbase + tile * 64;
#pragma unroll
    for (int i = 0; i < 4; ++i) {
      int e4 = i * 256 + t;            // 1024 float4s per tile
      int kk = e4 >> 4, n4 = e4 & 15;
      async_copy_b128((uint32_t)(uintptr_t)&Xl[buf][kk * 64 + n4 * 4],
                      x + xbase + (size_t)kk * M_DIM + n0 + n4 * 4);
    }
  };
  issueX(0, 0);

  // S_hat and t-vector stay resident across all NT tiles
#pragma unroll
  for (int i = 0; i < 4; ++i) {
    int e4 = i * 256 + t;
    *(float4*)&Sl[e4 * 4] = *(const float4*)(g_S + (size_t)b * 4096 + e4 * 4);
  }
  if (t < 64) tl[t] = g_t[b * 64 + t];

  const int ni = w & 3;
  const int mi0 = (w >> 2) * 2, mi1 = mi0 + 1;
  const int nrow = lane & 15;
  const int kb = (lane >> 4) * 2;
  const int mo = (lane >> 4) * 8;

  for (int tile = 0; tile < NT; ++tile) {
    const int cur = tile & 1;
    if (tile + 1 < NT) {
      issueX(tile + 1, cur ^ 1);       // overlap next tile fill with compute
      wait_asynccnt<4>();              // in-order: current tile landed
    } else {
      wait_asynccnt<0>();
    }
    __syncthreads();
    const float* X = Xl[cur];
    v8f acc0 = {}, acc1 = {};
    for (int k = 0; k < 64; k += 4) {
      v2f bb, a0, a1;
      bb.x = X[(k + kb) * 64 + ni * 16 + nrow];
      bb.y = X[(k + kb + 1) * 64 + ni * 16 + nrow];
      a0.x = Sl[(mi0 * 16 + nrow) * 64 + k + kb];
      a0.y = Sl[(mi0 * 16 + nrow) * 64 + k + kb + 1];
      a1.x = Sl[(mi1 * 16 + nrow) * 64 + k + kb];
      a1.y = Sl[(mi1 * 16 + nrow) * 64 + k + kb + 1];
      acc0 = wmma_f32_k4(a0, bb, acc0);
      acc1 = wmma_f32_k4(a1, bb, acc1);
    }
    const int col = nbase + tile * 64 + ni * 16 + nrow;
#pragma unroll
    for (int r = 0; r < 8; ++r) {
      int row0 = mi0 * 16 + r + mo;
      int row1 = mi1 * 16 + r + mo;
      out[xbase + (size_t)row0 * M_DIM + col] = acc0[r] - tl[row0];
      out[xbase + (size_t)row1 * M_DIM + col] = acc1[r] - tl[row1];
    }
    __syncthreads();                   // release buffer `cur` for refill
  }
}

// ---------------------------------------------------------------------------
extern "C" void kernel_launch(void* const* d_in, const int* in_sizes, int n_in,
                              void* d_out, int out_size, void* d_ws,
                              size_t ws_size, hipStream_t stream) {
  const float* x = (const float*)d_in[0];
  float* out = (float*)d_out;
  float* ws = (float*)d_ws;

  float* g_cov = ws;                       // 64*4096
  float* g_U   = g_cov + 64 * 4096;        // 64*4096
  float* g_L   = g_U + 64 * 4096;          // 64*64
  float* g_mu  = g_L + 64 * 64;            // 64*64
  float* g_t   = g_mu + 64 * 64;           // 64*64
  float* g_S   = g_t + 64 * 64;            // 64*4096
  int*   g_cnt = (int*)(g_S + 64 * 4096);  // 1
  (void)in_sizes; (void)n_in; (void)out_size; (void)ws_size;

  k_cov<<<BATCH, 256, 0, stream>>>(x, g_cov, g_mu);
  k_eig<<<BATCH, 64, 0, stream>>>(g_cov, g_U, g_L);
  k_counter<<<1, 64, 0, stream>>>(g_L, g_cnt);
  k_deflate<<<BATCH, 64, 0, stream>>>(g_cov, g_U, g_L, g_cnt, g_mu, g_S, g_t);
  k_zgemm<<<dim3(M_DIM / (64 * NT), BATCH), 256, 0, stream>>>(x, g_S, g_t, out);
}